// NUFFTLayerMultiChannel3D_Param_57801669869710
// MI455X (gfx1250) — compile-verified
//
#include <hip/hip_runtime.h>
#include <math.h>

// ---------- problem constants ----------
#define GG      32
#define G3      32768          // 32^3
#define NPTS    2048           // 32 lat * 64 lon
#define NCH     8
#define PI_F    3.14159265358979323846f
#define TWO_PI_F 6.28318530717958647692f
#define TAU_F   0.01171875f            // 12*(L/(2*pi*N))^2, L=2*pi, N=32
#define INV4TAU 21.333333333333332f    // 1/(4*TAU)
#define W32ANG  0.19634954084936207f   // 2*pi/32

// ---------- workspace layout (float offsets) ----------
#define OFF_GX     0u          // 2048*32
#define OFF_GY     65536u
#define OFF_GZ     131072u
#define OFF_HQ     196608u     // 32768 (filter, natural FFT order)
#define OFF_FPAD   229376u     // 16*2048   (A-matrix of GEMM1, rows 8..15 zero)
#define OFF_SUMMED 262144u     // 8*32768
#define OFF_CPLX0  524288u     // 8*32768 float2
#define OFF_CPLX1  1048576u    // 8*32768 float2
#define OFF_INVPAD 1572864u    // 16*32768  (A-matrix of GEMM2, rows 8..15 zero)
#define OFF_ENERGY 2097152u    // 8*2048
// total = 2,113,536 floats ~= 8.5 MB of d_ws

typedef float v2f __attribute__((ext_vector_type(2)));
typedef float v8f __attribute__((ext_vector_type(8)));

// periodic Gaussian spreading function
__device__ __forceinline__ float gfun(float d) {
    float b = d - TWO_PI_F;
    float c = d + TWO_PI_F;
    return expf(-d * d * INV4TAU) + expf(-b * b * INV4TAU) + expf(-c * c * INV4TAU);
}

// ---------- precompute separable kernel factors gx,gy,gz (2048 x 32 each) ----------
__global__ void precompute_g(float* __restrict__ gx, float* __restrict__ gy,
                             float* __restrict__ gz) {
    int tid = blockIdx.x * blockDim.x + threadIdx.x;   // 65536
    if (tid >= NPTS * GG) return;
    int n  = tid >> 5;
    int xi = tid & 31;
    int la_i = n >> 6;          // lat index (32)
    int lo_i = n & 63;          // lon index (64)
    float la = -0.5f * PI_F + PI_F * (float)la_i / 31.0f;
    float lo = TWO_PI_F * (float)lo_i / 63.0f;
    float cl = cosf(la);
    float px = cl * cosf(lo);
    float py = cl * sinf(lo);
    float pz = sinf(la);
    float xv = -PI_F + (TWO_PI_F / 32.0f) * (float)xi;  // xg grid point
    gx[tid] = gfun(px - xv);
    gy[tid] = gfun(py - xv);
    gz[tid] = gfun(pz - xv);
}

// ---------- filter in natural (unshifted) FFT order: H = deconv * total ----------
__global__ void precompute_filter(const float* __restrict__ amp, const float* __restrict__ shf,
                                  const float* __restrict__ bet, const float* __restrict__ hyp,
                                  float* __restrict__ hq) {
    int tid = blockIdx.x * blockDim.x + threadIdx.x;   // 32768
    if (tid >= G3) return;
    int i = tid >> 10, j = (tid >> 5) & 31, k = tid & 31;
    int si = (i + 16) & 31, sj = (j + 16) & 31, sk = (k + 16) & 31;  // ifftshift
    float kx = -16.0f + (32.0f / 31.0f) * (float)si;   // kg = linspace(-16,16,32)
    float ky = -16.0f + (32.0f / 31.0f) * (float)sj;
    float kz = -16.0f + (32.0f / 31.0f) * (float)sk;
    float k2 = kx * kx + ky * ky + kz * kz;
    float km = sqrtf(k2);
    float deconv = powf(PI_F / TAU_F, 1.5f) * expf(k2 * TAU_F);
    float total = 0.0f;
    #pragma unroll
    for (int c = 0; c < NCH; ++c) {
        float a  = amp[c];
        float s  = shf[c];
        float be = bet[c];
        float hy = hyp[c];
        float p1 = a / ((km + s) * (km + s) + a * a);
        float p2 = a / ((km - s) * (km - s) + a * a);
        float fl = k2 / (k2 + 400.0f * hy);
        total += be * (p1 + p2) * fl;
    }
    hq[tid] = deconv * total;
}

// ---------- init: padded A matrices + zeroed accumulators ----------
__global__ void init_bufs(const float* __restrict__ inp, float* __restrict__ fpad,
                          float* __restrict__ invpad, float* __restrict__ energy) {
    int tid = blockIdx.x * blockDim.x + threadIdx.x;   // 524288
    invpad[tid] = 0.0f;                                // 16*32768
    if (tid < 16 * NPTS) fpad[tid] = (tid < NCH * NPTS) ? inp[tid] : 0.0f;
    if (tid < NCH * NPTS) energy[tid] = 0.0f;
}

// ---------- GEMM1: summed[c,v] = sum_n fpad[c,n] * gx[n,x]gy[n,y]gz[n,z] ----------
// one wave per 16x16 tile; 2048 tiles cover N=32768, M rows 0..7 are channels.
// gx*gy (shared by all 16 z-columns of the tile) is staged once in LDS.
__global__ __launch_bounds__(32)
void gemm1_wmma(const float* __restrict__ fpad, const float* __restrict__ gx,
                const float* __restrict__ gy, const float* __restrict__ gz,
                float* __restrict__ summed) {
    __shared__ float gxy[NPTS];              // 8 KB
    const int v0   = blockIdx.x * 16;
    const int x    = v0 >> 10;
    const int y    = (v0 >> 5) & 31;
    const int z0   = v0 & 31;                // 0 or 16
    const int lane = threadIdx.x;
    const int m    = lane & 15;
    const int hi2  = (lane >> 4) << 1;       // 0 (lanes 0-15) or 2 (lanes 16-31)
    for (int n = lane; n < NPTS; n += 32)
        gxy[n] = gx[n * 32 + x] * gy[n * 32 + y];
    __syncthreads();
    const float*  gzp  = gz + z0 + m;                        // gz[n*32 + z0 + m]
    const float2* Arow = (const float2*)(fpad + m * NPTS);   // even-aligned
    __builtin_prefetch(fpad + m * NPTS, 0, 0);               // global_prefetch_b8
    v8f acc = {0.f, 0.f, 0.f, 0.f, 0.f, 0.f, 0.f, 0.f};
    for (int k0 = 0; k0 < NPTS; k0 += 4) {
        int ka = k0 + hi2;                                   // even
        float2 av = Arow[ka >> 1];                           // A[m][ka], A[m][ka+1]
        float2 gv = *(const float2*)&gxy[ka];                // ds_load_b64
        v2f A, B;
        A.x = av.x;  A.y = av.y;
        B.x = gv.x * gzp[ka * 32];
        B.y = gv.y * gzp[(ka + 1) * 32];
        acc = __builtin_amdgcn_wmma_f32_16x16x4_f32(false, A, false, B,
                                                    (short)0, acc, false, false);
    }
    if (lane < 16) {
        #pragma unroll
        for (int r = 0; r < NCH; ++r)         // C/D: VGPR r, lanes 0-15 -> row M=r
            summed[r * G3 + v0 + lane] = acc[r];
    }
}

// ---------- 32-pt DFT passes: 8 lines per 256-thread block, LDS-staged ----------
// line enumeration for axis of stride S: base = (L/S)*(S*32) + (L%S), elems base + a*S
__global__ __launch_bounds__(256)
void dft_r2c_z(const float* __restrict__ in, float2* __restrict__ out) {
    __shared__ float rline[8][32];
    __shared__ float cs[32], sn[32];
    const int l = threadIdx.x >> 5, t = threadIdx.x & 31;
    if (threadIdx.x < 32) {
        float s, c;
        __sincosf(-W32ANG * (float)t, &s, &c);   // forward twiddles
        cs[t] = c; sn[t] = s;
    }
    const int base = (blockIdx.x * 8 + l) * 32;
    rline[l][t] = in[base + t];
    __syncthreads();
    float re = 0.f, im = 0.f;
    #pragma unroll
    for (int u = 0; u < 32; ++u) {
        float v = rline[l][u];
        int tt = (t * u) & 31;
        re += v * cs[tt];
        im += v * sn[tt];
    }
    out[base + t] = make_float2(re, im);
}

__global__ __launch_bounds__(256)
void dft_c2c(const float2* __restrict__ in, float2* __restrict__ out,
             int stride, float sign, float scale) {
    __shared__ float2 line[8][32];
    __shared__ float cs[32], sn[32];
    const int l = threadIdx.x >> 5, t = threadIdx.x & 31;
    if (threadIdx.x < 32) {
        float s, c;
        __sincosf(sign * W32ANG * (float)t, &s, &c);   // sign folded into table
        cs[t] = c; sn[t] = s;
    }
    const int L = blockIdx.x * 8 + l;
    const int base = (L / stride) * (stride * 32) + (L % stride);
    line[l][t] = in[base + t * stride];
    __syncthreads();
    float re = 0.f, im = 0.f;
    #pragma unroll
    for (int u = 0; u < 32; ++u) {
        float2 v = line[l][u];
        int tt = (t * u) & 31;
        float c = cs[tt], s = sn[tt];
        re += v.x * c - v.y * s;
        im += v.x * s + v.y * c;
    }
    out[base + t * stride] = make_float2(re * scale, im * scale);
}

__global__ void apply_filter(float2* __restrict__ d, const float* __restrict__ hq) {
    int tid = blockIdx.x * blockDim.x + threadIdx.x;   // 262144
    float h = hq[tid & (G3 - 1)];
    float2 v = d[tid];
    d[tid] = make_float2(v.x * h, v.y * h);
}

__global__ __launch_bounds__(256)
void dft_c2r_z(const float2* __restrict__ in, float* __restrict__ outr) {
    __shared__ float2 line[8][32];
    __shared__ float cs[32], sn[32];
    const int l = threadIdx.x >> 5, t = threadIdx.x & 31;
    if (threadIdx.x < 32) {
        float s, c;
        __sincosf(W32ANG * (float)t, &s, &c);   // inverse twiddles
        cs[t] = c; sn[t] = s;
    }
    const int base = (blockIdx.x * 8 + l) * 32;
    line[l][t] = in[base + t];
    __syncthreads();
    float re = 0.f;
    #pragma unroll
    for (int u = 0; u < 32; ++u) {
        float2 v = line[l][u];
        int tt = (t * u) & 31;
        re += v.x * cs[tt] - v.y * sn[tt];
    }
    outr[base + t] = re * 0.03125f;   // last 1/32 of the 1/32768 ifftn normalization
}

// ---------- GEMM2: energy[c,n] = sum_v invpad[c,v] * K[n,v]; reduction split by x ----------
// per-tile factor rows staged in LDS (stride 33 to spread banks)
__global__ __launch_bounds__(32)
void gemm2_wmma(const float* __restrict__ invpad, const float* __restrict__ gx,
                const float* __restrict__ gy, const float* __restrict__ gz,
                float* __restrict__ energy) {
    __shared__ float hxy[16 * 33];           // gx[n,x]*gy[n,y] for tile rows
    __shared__ float hz [16 * 33];           // gz[n,z]          for tile rows
    const int n0   = blockIdx.x * 16;
    const int x    = blockIdx.y;             // v-slab: [x*1024, x*1024+1024)
    const int lane = threadIdx.x;
    const int m    = lane & 15;
    const int hi2  = (lane >> 4) << 1;
    for (int idx = lane; idx < 512; idx += 32) {
        int r = idx >> 5, q = idx & 31;      // r = tile row, q = y or z coord
        int nn = n0 + r;
        hxy[r * 33 + q] = gx[nn * 32 + x] * gy[nn * 32 + q];
        hz [r * 33 + q] = gz[nn * 32 + q];
    }
    __syncthreads();
    const int vbase = x << 10;
    const float2* Arow = (const float2*)(invpad + m * G3);
    const float*  hxyl = hxy + m * 33;
    const float*  hzl  = hz  + m * 33;
    __builtin_prefetch(invpad + m * G3 + vbase, 0, 0);   // global_prefetch_b8
    v8f acc = {0.f, 0.f, 0.f, 0.f, 0.f, 0.f, 0.f, 0.f};
    for (int vv = 0; vv < 1024; vv += 4) {
        int kv = vbase + vv + hi2;                       // even
        float2 av = Arow[kv >> 1];
        int z0k = kv & 31,       y0k = (kv >> 5) & 31;
        int z1k = (kv + 1) & 31, y1k = ((kv + 1) >> 5) & 31;
        v2f A, B;
        A.x = av.x;  A.y = av.y;
        B.x = hxyl[y0k] * hzl[z0k];
        B.y = hxyl[y1k] * hzl[z1k];
        acc = __builtin_amdgcn_wmma_f32_16x16x4_f32(false, A, false, B,
                                                    (short)0, acc, false, false);
    }
    if (lane < 16) {
        #pragma unroll
        for (int r = 0; r < NCH; ++r)
            atomicAdd(&energy[r * NPTS + n0 + m], acc[r]);   // global_atomic_add_f32
    }
}

// ---------- output relayout: pred.flat[n*8+c] = energy[c,n] ----------
__global__ void write_out(const float* __restrict__ energy, float* __restrict__ out) {
    int tid = blockIdx.x * blockDim.x + threadIdx.x;   // 16384
    if (tid >= NCH * NPTS) return;
    out[tid] = energy[(tid & 7) * NPTS + (tid >> 3)];
}

extern "C" void kernel_launch(void* const* d_in, const int* in_sizes, int n_in,
                              void* d_out, int out_size, void* d_ws, size_t ws_size,
                              hipStream_t stream) {
    const float* inp = (const float*)d_in[0];
    const float* amp = (const float*)d_in[1];
    const float* shf = (const float*)d_in[2];
    const float* bet = (const float*)d_in[3];
    const float* hyp = (const float*)d_in[4];
    float* ws = (float*)d_ws;

    float*  gx     = ws + OFF_GX;
    float*  gy     = ws + OFF_GY;
    float*  gz     = ws + OFF_GZ;
    float*  hq     = ws + OFF_HQ;
    float*  fpad   = ws + OFF_FPAD;
    float*  summed = ws + OFF_SUMMED;
    float2* c0     = (float2*)(ws + OFF_CPLX0);
    float2* c1     = (float2*)(ws + OFF_CPLX1);
    float*  invpad = ws + OFF_INVPAD;
    float*  energy = ws + OFF_ENERGY;
    float*  out    = (float*)d_out;

    precompute_g<<<256, 256, 0, stream>>>(gx, gy, gz);
    precompute_filter<<<128, 256, 0, stream>>>(amp, shf, bet, hyp, hq);
    init_bufs<<<2048, 256, 0, stream>>>(inp, fpad, invpad, energy);

    gemm1_wmma<<<2048, 32, 0, stream>>>(fpad, gx, gy, gz, summed);

    dft_r2c_z<<<1024, 256, 0, stream>>>(summed, c0);
    dft_c2c<<<1024, 256, 0, stream>>>(c0, c1, 32,   -1.0f, 1.0f);      // fwd y
    dft_c2c<<<1024, 256, 0, stream>>>(c1, c0, 1024, -1.0f, 1.0f);      // fwd x
    apply_filter<<<1024, 256, 0, stream>>>(c0, hq);
    dft_c2c<<<1024, 256, 0, stream>>>(c0, c1, 1024,  1.0f, 0.03125f);  // inv x
    dft_c2c<<<1024, 256, 0, stream>>>(c1, c0, 32,    1.0f, 0.03125f);  // inv y
    dft_c2r_z<<<1024, 256, 0, stream>>>(c0, invpad);                   // inv z -> real

    gemm2_wmma<<<dim3(128, 32), 32, 0, stream>>>(invpad, gx, gy, gz, energy);
    write_out<<<64, 256, 0, stream>>>(energy, out);
}